// DrugGNN_2310692405721
// MI455X (gfx1250) — compile-verified
//
#include <hip/hip_runtime.h>
#include <hip/hip_bf16.h>

#define HIDDEN 128

typedef float v2f __attribute__((ext_vector_type(2)));
typedef float v8f __attribute__((ext_vector_type(8)));

// ---------------------------------------------------------------- utilities

__global__ __launch_bounds__(256) void k_fill1(float* __restrict__ p, int n) {
  int i = blockIdx.x * 256 + threadIdx.x;
  if (i < n) p[i] = 1.0f;
}

__global__ __launch_bounds__(256) void k_zero(float* __restrict__ p, int n) {
  int i = blockIdx.x * 256 + threadIdx.x;
  if (i < n) p[i] = 0.0f;
}

__global__ __launch_bounds__(256) void k_degacc(const int* __restrict__ dst,
                                                float* __restrict__ deg, int E) {
  int e = blockIdx.x * 256 + threadIdx.x;
  if (e < E) atomicAdd(&deg[dst[e]], 1.0f);
}

__global__ __launch_bounds__(256) void k_rsqrt(float* __restrict__ p, int n) {
  int i = blockIdx.x * 256 + threadIdx.x;
  if (i < n) p[i] = rsqrtf(p[i]);
}

// ---------------------------------------------------------------- fused WMMA GEMM
// hw[N,128]  = A @ W            (A = relu(prev activations), materialized in regs)
// agg[N,128] = dinv^2*hw + b    (self-loop + bias seed for the edge scatter)
//
// f32 WMMA 16x16x4 per ISA 7.12.2:
//   A-frag 16x4 : lane L -> M=L&15; VGPR0=K(khalf), VGPR1=K+1; khalf=(L>=16)?2:0
//   B-frag 4x16 : VGPR0 lanes0-15 -> K=khalf, N=lane&15; lanes16-31 -> K=khalf(=2)
//   C-frag 16x16: VGPR i -> M=i+(lane>=16?8:0), N=lane&15
// Block = 256 thr = 8 waves; block owns 16 rows, wave w owns cols [16w,16w+16).
// EXEC stays all-1s: row clamp is a v_cndmask, stores use a wave-uniform guard.
template <bool RANK1>
__global__ __launch_bounds__(256) void k_gemm_fused(
    const float* __restrict__ Ain,   // RANK1: x[N] ; else aggPrev[N,128]
    const float* __restrict__ Wlin,  // RANK1: W_in[128] ; else unused
    const float* __restrict__ blin,  // RANK1: b_in[128] ; else unused
    const float* __restrict__ W,     // [128,128]
    const float* __restrict__ bconv, // [128]
    const float* __restrict__ dinv,  // [N]
    float* __restrict__ hw,          // out [N,128]
    float* __restrict__ agg,         // out [N,128]
    int n_rows) {
  const int lane = threadIdx.x & 31;
  const int wave = threadIdx.x >> 5;
  const int row_base = blockIdx.x << 4;
  const int col_base = wave << 4;
  const int m = lane & 15;
  const int khalf = (lane >> 4) << 1;         // 0 or 2
  int row = row_base + m;
  row = (row < n_rows) ? row : (n_rows - 1);  // clamp, no EXEC divergence
  const int n = col_base + m;

  const float* __restrict__ wcol = W + (size_t)khalf * HIDDEN + n;
  const float* __restrict__ arow =
      RANK1 ? nullptr : (Ain + (size_t)row * HIDDEN + khalf);
  const float xr = RANK1 ? Ain[row] : 0.0f;

  v8f acc = {};
#pragma unroll 8
  for (int kk = 0; kk < HIDDEN; kk += 4) {
    v2f a, b;
    if (RANK1) {
      const int k0 = kk + khalf;
      a.x = fmaxf(fmaf(xr, Wlin[k0], blin[k0]), 0.0f);
      a.y = fmaxf(fmaf(xr, Wlin[k0 + 1], blin[k0 + 1]), 0.0f);
    } else {
      a.x = fmaxf(arow[kk], 0.0f);       // fused ReLU of previous layer
      a.y = fmaxf(arow[kk + 1], 0.0f);
    }
    b.x = wcol[(size_t)kk * HIDDEN];
    b.y = wcol[(size_t)(kk + 1) * HIDDEN];
    acc = __builtin_amdgcn_wmma_f32_16x16x4_f32(false, a, false, b,
                                                (short)0, acc, false, false);
  }

  const float bb = bconv[n];
  const int r0 = row_base + ((lane >> 4) << 3);
  if (row_base + 16 <= n_rows) {             // wave-uniform fast path
#pragma unroll
    for (int i = 0; i < 8; ++i) {
      const int r = r0 + i;
      const float v = acc[i];
      const float d = dinv[r];
      hw[(size_t)r * HIDDEN + n] = v;
      agg[(size_t)r * HIDDEN + n] = fmaf(d * d, v, bb);
    }
  } else {
#pragma unroll
    for (int i = 0; i < 8; ++i) {
      const int r = r0 + i;
      if (r < n_rows) {
        const float v = acc[i];
        const float d = dinv[r];
        hw[(size_t)r * HIDDEN + n] = v;
        agg[(size_t)r * HIDDEN + n] = fmaf(d * d, v, bb);
      }
    }
  }
}

// agg[dst,f] += dinv[src]*dinv[dst]*hw[src,f] ; 64 lanes x float2 per edge
__global__ __launch_bounds__(256) void k_scatter(const float* __restrict__ hw,
                                                 const float* __restrict__ dinv,
                                                 const int* __restrict__ src,
                                                 const int* __restrict__ dst,
                                                 float* __restrict__ agg, int E) {
  int t = blockIdx.x * 256 + threadIdx.x;
  if (t < E * (HIDDEN / 2)) {
    int e = t >> 6, f2 = (t & 63) << 1;
    int s = src[e], d = dst[e];
    float w = dinv[s] * dinv[d];
    const v2f hv = *(const v2f*)(hw + (size_t)s * HIDDEN + f2);
    float* ap = agg + (size_t)d * HIDDEN + f2;
    atomicAdd(ap, w * hv.x);
    atomicAdd(ap + 1, w * hv.y);
  }
}

// pool[batch[n],f] += relu(agg[n,f]) ; cnt[g] += 1 once per node
__global__ __launch_bounds__(256) void k_pool(const float* __restrict__ aggf,
                                              const int* __restrict__ batch,
                                              float* __restrict__ pool,
                                              float* __restrict__ cnt, int n_nodes) {
  int t = blockIdx.x * 256 + threadIdx.x;
  if (t < n_nodes * (HIDDEN / 2)) {
    int nn = t >> 6, f2 = (t & 63) << 1;
    int g = batch[nn];
    const v2f hv = *(const v2f*)(aggf + (size_t)nn * HIDDEN + f2);
    float* pp = pool + (size_t)g * HIDDEN + f2;
    atomicAdd(pp, fmaxf(hv.x, 0.0f));
    atomicAdd(pp + 1, fmaxf(hv.y, 0.0f));
    if ((t & 63) == 0) atomicAdd(&cnt[g], 1.0f);
  }
}

// out[g,f] = relu( (pool[g,:]/max(cnt[g],1)) @ W_out + b_out )  via f32 WMMA
__global__ __launch_bounds__(256) void k_outgemm(const float* __restrict__ pool,
                                                 const float* __restrict__ cnt,
                                                 const float* __restrict__ W,
                                                 const float* __restrict__ bias,
                                                 float* __restrict__ out, int n_rows) {
  const int lane = threadIdx.x & 31;
  const int wave = threadIdx.x >> 5;
  const int row_base = blockIdx.x << 4;
  const int col_base = wave << 4;
  const int m = lane & 15;
  const int khalf = (lane >> 4) << 1;
  int row = row_base + m;
  row = (row < n_rows) ? row : (n_rows - 1);
  const int n = col_base + m;

  const float invc = 1.0f / fmaxf(cnt[row], 1.0f);
  const float* __restrict__ arow = pool + (size_t)row * HIDDEN + khalf;
  const float* __restrict__ wcol = W + (size_t)khalf * HIDDEN + n;

  v8f acc = {};
#pragma unroll 8
  for (int kk = 0; kk < HIDDEN; kk += 4) {
    v2f a, b;
    a.x = arow[kk] * invc;
    a.y = arow[kk + 1] * invc;
    b.x = wcol[(size_t)kk * HIDDEN];
    b.y = wcol[(size_t)(kk + 1) * HIDDEN];
    acc = __builtin_amdgcn_wmma_f32_16x16x4_f32(false, a, false, b,
                                                (short)0, acc, false, false);
  }

  const float bb = bias[n];
  const int r0 = row_base + ((lane >> 4) << 3);
  if (row_base + 16 <= n_rows) {
#pragma unroll
    for (int i = 0; i < 8; ++i)
      out[(size_t)(r0 + i) * HIDDEN + n] = fmaxf(acc[i] + bb, 0.0f);
  } else {
#pragma unroll
    for (int i = 0; i < 8; ++i) {
      const int r = r0 + i;
      if (r < n_rows) out[(size_t)r * HIDDEN + n] = fmaxf(acc[i] + bb, 0.0f);
    }
  }
}

// ---------------------------------------------------------------- launcher

static inline int cdiv(long long a, long long b) { return (int)((a + b - 1) / b); }

extern "C" void kernel_launch(void* const* d_in, const int* in_sizes, int n_in,
                              void* d_out, int out_size, void* d_ws, size_t ws_size,
                              hipStream_t stream) {
  const float* x       = (const float*)d_in[0];
  const int*   edge    = (const int*)d_in[1];   // [2,E]
  const int*   batch   = (const int*)d_in[2];   // [N]
  const float* W_in    = (const float*)d_in[4]; // [1,128]
  const float* b_in    = (const float*)d_in[5];
  const float* W_convs = (const float*)d_in[6]; // [L,128,128]
  const float* b_convs = (const float*)d_in[7]; // [L,128]
  const float* W_out   = (const float*)d_in[8];
  const float* b_out   = (const float*)d_in[9];
  float* out = (float*)d_out;

  const int N = in_sizes[0];               // x is [N,1]
  const int E = in_sizes[1] / 2;
  const int L = in_sizes[6] / (HIDDEN * HIDDEN);
  const int G = out_size / HIDDEN;

  const int* src = edge;
  const int* dst = edge + E;

  // workspace carve-up (floats)
  float* ws = (float*)d_ws;
  size_t o = 0;
  float* dinv = ws + o; o += (size_t)((N + 63) & ~63);
  float* hw   = ws + o; o += (size_t)N * HIDDEN;
  float* agg0 = ws + o; o += (size_t)N * HIDDEN;
  float* agg1 = ws + o; o += (size_t)N * HIDDEN;
  float* pool = ws + o; o += (size_t)G * HIDDEN;
  float* cnt  = ws + o; o += (size_t)G;
  (void)ws_size; (void)n_in;

  const long long NH2 = (long long)N * (HIDDEN / 2);
  const long long EH2 = (long long)E * (HIDDEN / 2);

  // degrees -> dinv = (in_deg + 1)^-1/2
  k_fill1 <<<cdiv(N, 256), 256, 0, stream>>>(dinv, N);
  k_degacc<<<cdiv(E, 256), 256, 0, stream>>>(dst, dinv, E);
  k_rsqrt <<<cdiv(N, 256), 256, 0, stream>>>(dinv, N);

  // conv layers: layer 0 fuses the rank-1 input projection into the A-fragment
  float* aggIn = nullptr;  // not used by layer 0
  float* aggOut = agg0;
  for (int l = 0; l < L; ++l) {
    const float* Wl = W_convs + (size_t)l * HIDDEN * HIDDEN;
    const float* bl = b_convs + (size_t)l * HIDDEN;
    if (l == 0) {
      k_gemm_fused<true><<<cdiv(N, 16), 256, 0, stream>>>(
          x, W_in, b_in, Wl, bl, dinv, hw, aggOut, N);
    } else {
      k_gemm_fused<false><<<cdiv(N, 16), 256, 0, stream>>>(
          aggIn, nullptr, nullptr, Wl, bl, dinv, hw, aggOut, N);
    }
    k_scatter<<<cdiv(EH2, 256), 256, 0, stream>>>(hw, dinv, src, dst, aggOut, E);
    aggIn = aggOut;
    aggOut = (aggOut == agg0) ? agg1 : agg0;
  }
  float* aggFinal = aggIn;  // result of last layer (pre-ReLU; pool applies it)

  // mean pool + output layer (pool & cnt contiguous -> one zero pass)
  k_zero   <<<cdiv((long long)G * HIDDEN + G, 256), 256, 0, stream>>>(pool, G * HIDDEN + G);
  k_pool   <<<cdiv(NH2, 256), 256, 0, stream>>>(aggFinal, batch, pool, cnt, N);
  k_outgemm<<<cdiv(G, 16), 256, 0, stream>>>(pool, cnt, W_out, b_out, out, G);
}